// PDEsimpleModel_2018634629585
// MI455X (gfx1250) — compile-verified
//
#include <hip/hip_runtime.h>
#include <stdint.h>

// Problem constants (match the JAX reference).
#define NSTEPS 50
#define ALPHA  0.1f
#define NB     8      // batch
#define NN     2048   // stencil axis (periodic)
#define ND     128    // inner (independent) axis
#define DT     4      // d-values per block -> one 16B float4 "row" per n
#define NT     256    // threads per block (8 wave32)
#define RPT    8      // consecutive rows per thread (2048 / 256)

// Padded LDS row index (float4 units): one extra float4 per 8-row chunk.
// dword base for lane t, row 8t+i is 36t+4i -> 2 lanes per 4-bank group,
// each bank hit exactly twice per b128 wave op = minimum 2 bank-cycles.
__device__ __forceinline__ unsigned idx4(int r) {
    return (unsigned)(r + (r >> 3));
}

// ---- CDNA5 async LDS<->global ops (ASYNCcnt-tracked) via inline asm ----
__device__ __forceinline__ void async_load_b128(unsigned lds_off, const void* gaddr) {
    asm volatile("global_load_async_to_lds_b128 %0, %1, off"
                 :: "v"(lds_off), "v"((unsigned long long)(uintptr_t)gaddr)
                 : "memory");
}
__device__ __forceinline__ void async_store_b128(void* gaddr, unsigned lds_off) {
    asm volatile("global_store_async_from_lds_b128 %0, %1, off"
                 :: "v"((unsigned long long)(uintptr_t)gaddr), "v"(lds_off)
                 : "memory");
}
__device__ __forceinline__ void wait_async_0() {
    asm volatile("s_wait_asynccnt 0" ::: "memory");
}
__device__ __forceinline__ void wait_async_8() {
    asm volatile("s_wait_asynccnt 8" ::: "memory");
}

// One block owns (b, dbase..dbase+3): 8*32 = 256 blocks.
// State tile (2048 float4 rows, padded = 36 KB) lives in LDS, double buffered.
// All 50 steps run in-LDS; each step's result is async-stored to HBM,
// overlapping the next step's compute (HBM floor ~= 428MB / 23.3TB/s ~ 19us).
__global__ __launch_bounds__(NT) void pde_diffuse_kernel(const float* __restrict__ in,
                                                         float* __restrict__ out) {
    __shared__ __align__(16) float4 bufA[NN + NN / 8];   // 2304 float4 = 36 KB
    __shared__ __align__(16) float4 bufB[NN + NN / 8];

    const int t     = threadIdx.x;
    const int b     = blockIdx.x >> 5;            // 32 d-tiles per batch
    const int dbase = (blockIdx.x & 31) * DT;
    const int r0    = t * RPT;                    // thread owns rows [r0, r0+8)

    // Low 32 bits of a flat shared pointer == wave-relative LDS byte offset.
    const unsigned ldsA = (unsigned)(uintptr_t)bufA;
    const unsigned ldsB = (unsigned)(uintptr_t)bufB;

    // ---- initial fill: HBM -> LDS (async), rows owned per-thread ----
    const float* inBase = in + (size_t)b * NN * ND + dbase;
    #pragma unroll
    for (int i = 0; i < RPT; ++i) {
        const int r = r0 + i;
        async_load_b128(ldsA + idx4(r) * 16u, inBase + (size_t)r * ND);
    }
    wait_async_0();
    __syncthreads();

    // ---- emit step 0 (the initial state) ----
    {
        float* out0 = out + (size_t)b * NN * ND + dbase;
        #pragma unroll
        for (int i = 0; i < RPT; ++i) {
            const int r = r0 + i;
            async_store_b128(out0 + (size_t)r * ND, ldsA + idx4(r) * 16u);
        }
    }

    const float4* cur = bufA;
    float4*       nxt = bufB;
    unsigned ldsCur = ldsA, ldsNxt = ldsB;
    (void)ldsCur;

    for (int step = 1; step <= NSTEPS; ++step) {
        // Drain all but the newest 8 store instructions (this wave): the
        // buffer about to be overwritten (stored 2 steps ago by this same
        // wave; stores complete in order) is guaranteed done, while the
        // previous step's stores keep streaming under the compute below.
        wait_async_8();

        // stencil: s' = s + a*(s[n-1] + s[n+1] - 2s), periodic in n.
        // float4 rows + register rolling: 10 b128 loads, 8 b128 stores.
        float4 p = cur[idx4((r0 - 1) & (NN - 1))];
        float4 c = cur[idx4(r0)];
        #pragma unroll
        for (int i = 0; i < RPT; ++i) {
            const int r = r0 + i;
            const float4 nx = cur[idx4((r + 1) & (NN - 1))];
            float4 o;
            o.x = fmaf(ALPHA, (p.x + nx.x) - 2.0f * c.x, c.x);
            o.y = fmaf(ALPHA, (p.y + nx.y) - 2.0f * c.y, c.y);
            o.z = fmaf(ALPHA, (p.z + nx.z) - 2.0f * c.z, c.z);
            o.w = fmaf(ALPHA, (p.w + nx.w) - 2.0f * c.w, c.w);
            nxt[idx4(r)] = o;
            p = c;
            c = nx;
        }
        __syncthreads();   // nxt fully written & visible before async stores / next reads

        // async store this step's tile: LDS -> out[step]
        float* outS = out + ((size_t)(step * NB + b) * NN) * ND + dbase;
        #pragma unroll
        for (int i = 0; i < RPT; ++i) {
            const int r = r0 + i;
            async_store_b128(outS + (size_t)r * ND, ldsNxt + idx4(r) * 16u);
        }

        // ping-pong
        const float4* tc = cur; cur = nxt; nxt = (float4*)tc;
        const unsigned tl = ldsCur; ldsCur = ldsNxt; ldsNxt = tl;
    }

    wait_async_0();  // drain before wave teardown (S_ENDPGM also waits-idle)
}

extern "C" void kernel_launch(void* const* d_in, const int* in_sizes, int n_in,
                              void* d_out, int out_size, void* d_ws, size_t ws_size,
                              hipStream_t stream) {
    (void)in_sizes; (void)n_in; (void)out_size; (void)d_ws; (void)ws_size;
    const float* in  = (const float*)d_in[0];
    float*       out = (float*)d_out;
    pde_diffuse_kernel<<<dim3(NB * (ND / DT)), dim3(NT), 0, stream>>>(in, out);
}